// INGPHashEncoder_56014963475028
// MI455X (gfx1250) — compile-verified
//
#include <hip/hip_runtime.h>
#include <hip/hip_bf16.h>

#define N_LEVELS   16
#define LOG2_T     19
#define TABLE_SIZE (1u << LOG2_T)
#define TMASK      (TABLE_SIZE - 1u)
#define P1         2654435761u
#define P2         805459861u

typedef float v2f __attribute__((ext_vector_type(2)));

// floor(16 * 2^(l/3)) for l = 0..15 (matches RESOLUTIONS in the reference)
__constant__ float c_res[N_LEVELS] = {
    16.0f, 20.0f, 25.0f, 32.0f, 40.0f, 50.0f, 64.0f, 80.0f,
    101.0f, 128.0f, 161.0f, 203.0f, 256.0f, 322.0f, 406.0f, 512.0f
};

// One thread per (point, level). lane = level + 16*(point parity):
// a wave32 covers all 16 levels of two consecutive points, so the wave's
// output region is one contiguous 256-byte block of float2 NT stores.
__global__ __launch_bounds__(256)
void ingp_hash_encode(const float* __restrict__ x,
                      const float* __restrict__ emb,
                      float* __restrict__ out,
                      unsigned npts)
{
    const unsigned tid   = blockIdx.x * blockDim.x + threadIdx.x;
    const unsigned level = tid & (N_LEVELS - 1);
    const unsigned point = tid >> 4;
    if (point >= npts) return;

    // Stream-prefetch x ~512 points ahead. TH=0 / SYS scope -> speculative
    // prefetch: an out-of-range address is silently dropped, so no bounds
    // check (and no exec-mask branch) is needed.
    __builtin_prefetch(x + 3u * (point + 512u), 0, 0);

    // ---- load & clip point (16 lanes share the same 12 bytes -> L0 hit) ----
    const float px0 = x[3u * point + 0u];
    const float py0 = x[3u * point + 1u];
    const float pz0 = x[3u * point + 2u];
    const float px = fminf(fmaxf(px0, -1.0f), 1.0f);
    const float py = fminf(fmaxf(py0, -1.0f), 1.0f);
    const float pz = fminf(fmaxf(pz0, -1.0f), 1.0f);

    // ---- per-level voxel coords + trilinear weights ----
    const float res  = c_res[level];
    const float g    = 2.0f / res;     // grid spacing
    const float invg = 0.5f * res;     // 1/grid

    const float fx = floorf((px + 1.0f) * invg);
    const float fy = floorf((py + 1.0f) * invg);
    const float fz = floorf((pz + 1.0f) * invg);

    const float wx = (px - (fx * g - 1.0f)) * invg;
    const float wy = (py - (fy * g - 1.0f)) * invg;
    const float wz = (pz - (fz * g - 1.0f)) * invg;

    // ---- spatial hash of the 8 corners (prime mul folded to adds) ----
    const unsigned ux = (unsigned)(int)fx;
    const unsigned uy = (unsigned)(int)fy;
    const unsigned uz = (unsigned)(int)fz;

    const unsigned hx0 = ux;             const unsigned hx1 = ux + 1u;   // prime 1
    const unsigned hy0 = uy * P1;        const unsigned hy1 = hy0 + P1;
    const unsigned hz0 = uz * P2;        const unsigned hz1 = hz0 + P2;

    // ---- 8 independent float2 gathers from the L2-resident table ----
    const v2f* __restrict__ t2 =
        (const v2f*)(emb + (size_t)level * (size_t)(TABLE_SIZE * 2u));

    const v2f e000 = t2[(hx0 ^ hy0 ^ hz0) & TMASK];
    const v2f e001 = t2[(hx0 ^ hy0 ^ hz1) & TMASK];
    const v2f e010 = t2[(hx0 ^ hy1 ^ hz0) & TMASK];
    const v2f e011 = t2[(hx0 ^ hy1 ^ hz1) & TMASK];
    const v2f e100 = t2[(hx1 ^ hy0 ^ hz0) & TMASK];
    const v2f e101 = t2[(hx1 ^ hy0 ^ hz1) & TMASK];
    const v2f e110 = t2[(hx1 ^ hy1 ^ hz0) & TMASK];
    const v2f e111 = t2[(hx1 ^ hy1 ^ hz1) & TMASK];

    // ---- trilinear blend (nested lerps == product-of-weights form) ----
    const float ox = 1.0f - wx, oy = 1.0f - wy, oz = 1.0f - wz;

    const v2f m00 = e000 * oz + e001 * wz;
    const v2f m01 = e010 * oz + e011 * wz;
    const v2f m10 = e100 * oz + e101 * wz;
    const v2f m11 = e110 * oz + e111 * wz;

    const v2f my0 = m00 * oy + m01 * wy;
    const v2f my1 = m10 * oy + m11 * wy;

    const v2f r = my0 * ox + my1 * wx;

    // ---- non-temporal store: output is write-once, keep table in L2 ----
    v2f* dst = (v2f*)(out + (size_t)point * (N_LEVELS * 2u) + level * 2u);
    __builtin_nontemporal_store(r, dst);
}

extern "C" void kernel_launch(void* const* d_in, const int* in_sizes, int n_in,
                              void* d_out, int out_size, void* d_ws, size_t ws_size,
                              hipStream_t stream)
{
    const float* x   = (const float*)d_in[0];   // (N, 3) float32
    const float* emb = (const float*)d_in[1];   // (16, 2^19, 2) float32
    float*       out = (float*)d_out;           // (N, 32) float32

    const unsigned npts    = (unsigned)(in_sizes[0] / 3);
    const unsigned threads = npts * N_LEVELS;
    const unsigned block   = 256;
    const unsigned grid    = (threads + block - 1) / block;

    ingp_hash_encode<<<grid, block, 0, stream>>>(x, emb, out, npts);
}